// SAGE_BinaryClassifier_10033043603760
// MI455X (gfx1250) — compile-verified
//
#include <hip/hip_runtime.h>

#define N_NODES 100000
#define N_EDGES 1600000
#define N_FEAT  64
#define HIDDEN  16
#define N_TILES (N_NODES / 16)   // 6250 exactly

typedef __attribute__((ext_vector_type(2))) float v2f;
typedef __attribute__((ext_vector_type(8))) float v8f;

// ---------------------------------------------------------------- zero scratch
__global__ void zero_f32(float* __restrict__ p, int n) {
    int i = blockIdx.x * blockDim.x + threadIdx.x;
    if (i < n) p[i] = 0.0f;
}

// ---------------------------------------------------------------- edge mask
// ewm = w0*As[0,e] + w1*As[1,e];  mask = (ewm != 0) ? 1 : 0
__global__ void edge_mask_kernel(const float* __restrict__ As,
                                 const float* __restrict__ w,
                                 float* __restrict__ mask) {
    int e = blockIdx.x * blockDim.x + threadIdx.x;
    if (e < N_EDGES) {
        float ewm = w[0] * As[e] + w[1] * As[N_EDGES + e];
        mask[e] = (ewm != 0.0f) ? 1.0f : 0.0f;
    }
}

// ---------------------------------------------------------------- layer-1 scatter
// one thread per (edge, feature); a wave32 covers 32 consecutive features of one
// edge -> 128B coalesced gather from x[src] and coalesced f32 atomics to agg1[dst]
__global__ void scatter_add_64(const float* __restrict__ x,
                               const long long* __restrict__ ei,
                               const float* __restrict__ mask,
                               float* __restrict__ agg1) {
    int gid = blockIdx.x * blockDim.x + threadIdx.x;
    if (gid >= N_EDGES * N_FEAT) return;
    int e = gid >> 6;
    int f = gid & 63;
    float m = mask[e];
    if (m != 0.0f) {
        int src = (int)ei[e];
        int dst = (int)ei[N_EDGES + e];
        float v = x[src * N_FEAT + f] * m;
        __hip_atomic_fetch_add(&agg1[dst * N_FEAT + f], v,
                               __ATOMIC_RELAXED, __HIP_MEMORY_SCOPE_AGENT);
    }
}

// ---------------------------------------------------------------- layer-1 fused GEMM
// h = relu(agg1 @ Wl1 + x @ Wr1 + bl1)  via V_WMMA_F32_16X16X4_F32
// one wave32 per 16x16 output tile; K=64 stepped by 4, two A/B streams into one C.
//
// Assumed f32 WMMA VGPR layouts (ISA 7.12.2):
//   A 16x4 : lane<16 holds row M=lane, K={k0,k0+1}; lane>=16 holds M=lane-16, K={k0+2,k0+3}
//   B 4x16 : lane<16 holds col N=lane, rows K={k0,k0+1}; lane>=16 col N=lane-16, K={k0+2,k0+3}
//   C 16x16: VGPR v -> M = v + (lane>=16 ? 8 : 0), N = lane&15
__global__ __launch_bounds__(256) void fused_layer1(
        const float* __restrict__ agg1, const float* __restrict__ x,
        const float* __restrict__ Wl1,  const float* __restrict__ Wr1,
        const float* __restrict__ bl1,  float* __restrict__ h) {
    const int wave = threadIdx.x >> 5;
    const int lane = threadIdx.x & 31;
    const int tile = blockIdx.x * 8 + wave;
    if (tile >= N_TILES) return;               // wave-uniform: EXEC all-1s inside

    const int row = tile * 16 + (lane & 15);   // A row this lane supplies
    const int kh  = (lane >> 4) * 2;           // K sub-offset: 0 or 2
    const int col = lane & 15;                 // B/C column this lane supplies

    v8f c = {};
    #pragma unroll
    for (int k0 = 0; k0 < N_FEAT; k0 += 4) {
        v2f a = *(const v2f*)(agg1 + row * N_FEAT + k0 + kh);   // 8B aligned
        v2f b;
        b.x = Wl1[(k0 + kh)     * HIDDEN + col];
        b.y = Wl1[(k0 + kh + 1) * HIDDEN + col];
        c = __builtin_amdgcn_wmma_f32_16x16x4_f32(false, a, false, b,
                                                  (short)0, c, false, false);
    }
    #pragma unroll
    for (int k0 = 0; k0 < N_FEAT; k0 += 4) {
        v2f a = *(const v2f*)(x + row * N_FEAT + k0 + kh);
        v2f b;
        b.x = Wr1[(k0 + kh)     * HIDDEN + col];
        b.y = Wr1[(k0 + kh + 1) * HIDDEN + col];
        c = __builtin_amdgcn_wmma_f32_16x16x4_f32(false, a, false, b,
                                                  (short)0, c, false, false);
    }

    const float bias = bl1[col];
    const int   mhi  = (lane >> 4) * 8;
    #pragma unroll
    for (int v = 0; v < 8; ++v) {
        float val = c[v] + bias;
        h[(tile * 16 + v + mhi) * HIDDEN + col] = fmaxf(val, 0.0f);
    }
}

// ---------------------------------------------------------------- layer-2 scatter
// one thread per (edge, hidden-feature); wave32 covers 2 edges x 16 features
__global__ void scatter_add_16(const float* __restrict__ h,
                               const long long* __restrict__ ei,
                               const float* __restrict__ mask,
                               float* __restrict__ agg2) {
    int gid = blockIdx.x * blockDim.x + threadIdx.x;
    if (gid >= N_EDGES * HIDDEN) return;
    int e = gid >> 4;
    int f = gid & 15;
    float m = mask[e];
    if (m != 0.0f) {
        int src = (int)ei[e];
        int dst = (int)ei[N_EDGES + e];
        float v = h[src * HIDDEN + f] * m;
        __hip_atomic_fetch_add(&agg2[dst * HIDDEN + f], v,
                               __ATOMIC_RELAXED, __HIP_MEMORY_SCOPE_AGENT);
    }
}

// ---------------------------------------------------------------- output head
// out[i] = agg2[i,:]·Wl2 + bl2 + h[i,:]·Wr2   (length-16 dots, one thread/node)
__global__ void output_head(const float* __restrict__ agg2,
                            const float* __restrict__ h,
                            const float* __restrict__ Wl2,
                            const float* __restrict__ bl2,
                            const float* __restrict__ Wr2,
                            float* __restrict__ out) {
    int i = blockIdx.x * blockDim.x + threadIdx.x;
    if (i >= N_NODES) return;
    float acc = bl2[0];
    #pragma unroll
    for (int k = 0; k < HIDDEN; ++k) {
        acc = fmaf(agg2[i * HIDDEN + k], Wl2[k], acc);
        acc = fmaf(h[i * HIDDEN + k],    Wr2[k], acc);
    }
    out[i] = acc;
}

// ---------------------------------------------------------------- launcher
extern "C" void kernel_launch(void* const* d_in, const int* in_sizes, int n_in,
                              void* d_out, int out_size, void* d_ws, size_t ws_size,
                              hipStream_t stream) {
    const float*     x   = (const float*)d_in[0];       // [N, 64]
    const long long* ei  = (const long long*)d_in[1];   // [2, E] int64
    const float*     As  = (const float*)d_in[2];       // [2, E]
    const float*     wsw = (const float*)d_in[3];       // [2]
    const float*     Wl1 = (const float*)d_in[4];       // [64, 16]
    const float*     bl1 = (const float*)d_in[5];       // [16]
    const float*     Wr1 = (const float*)d_in[6];       // [64, 16]
    const float*     Wl2 = (const float*)d_in[7];       // [16, 1]
    const float*     bl2 = (const float*)d_in[8];       // [1]
    const float*     Wr2 = (const float*)d_in[9];       // [16, 1]
    float* out = (float*)d_out;                         // [N, 1]

    // workspace layout (floats): agg1[N*64] | agg2[N*16] | h[N*16] | mask[E]
    float* agg1 = (float*)d_ws;
    float* agg2 = agg1 + (size_t)N_NODES * N_FEAT;
    float* h    = agg2 + (size_t)N_NODES * HIDDEN;
    float* mask = h    + (size_t)N_NODES * HIDDEN;

    const int B = 256;

    // zero accumulators (agg1 and agg2 are contiguous: N*(64+16) floats)
    int nz = N_NODES * (N_FEAT + HIDDEN);
    zero_f32<<<(nz + B - 1) / B, B, 0, stream>>>(agg1, nz);

    edge_mask_kernel<<<(N_EDGES + B - 1) / B, B, 0, stream>>>(As, wsw, mask);

    scatter_add_64<<<(N_EDGES * N_FEAT + B - 1) / B, B, 0, stream>>>(x, ei, mask, agg1);

    fused_layer1<<<(N_TILES + 7) / 8, B, 0, stream>>>(agg1, x, Wl1, Wr1, bl1, h);

    scatter_add_16<<<(N_EDGES * HIDDEN + B - 1) / B, B, 0, stream>>>(h, ei, mask, agg2);

    output_head<<<(N_NODES + B - 1) / B, B, 0, stream>>>(agg2, h, Wl2, bl2, Wr2, out);
}